// DCRF_29343216566879
// MI455X (gfx1250) — compile-verified
//
#include <hip/hip_runtime.h>
#include <math.h>

// ---------------------------------------------------------------------------
// DenseCRF mean-field, separable-pairwise formulation.
//   PP[b,i,j] = a_i a_j (1 - u_i.u_j) Wsym[i,j]
//   E_i = a_i [ (Wsym s)_i - ux_i (Wsym s*ux)_i - uy_i (Wsym s*uy)_i ]
// => per iteration one skinny fp32 GEMM: Y(4096x32) = Wsym(4096x4096) @ V(4096x32)
// executed with V_WMMA_F32_16X16X4_F32 (full fp32 precision, memory-bound).
//
// Tiling: one wave owns a 16-row slab of Wsym and BOTH 16-col output tiles
// (one A-fragment load feeds two WMMAs) -> Wsym streamed exactly once per
// iteration (64 MB, L2-resident).  Split-K=4 gives 1024 waves; fp32 partials
// are reduced in the cheap per-point kernels.  V is staged in LDS with a
// K-pair-interleaved layout so each B fragment is ONE aligned ds_load_b64
// (no register re-packing), banks disjoint across half-waves (PSTR=96).
// ---------------------------------------------------------------------------

typedef float v2f __attribute__((ext_vector_type(2)));
typedef float v8f __attribute__((ext_vector_type(8)));

#define BATCH  8
#define NPTS   4096
#define NC     32          // padded V/Y columns (24 used: 3 per batch)
#define KC     256         // K-chunk of V staged in LDS
#define PSTR   96          // floats per K-pair row in LDS (64 data + 32 pad)
#define ITERS  10
#define SPLITK 4
#define KSPAN  (NPTS / SPLITK)   // 1024

// ---- per-point precompute: a=exp(-|f|^2/2), unit dir, L=logits --------------
__global__ void crf_prep(const float* __restrict__ dp,
                         const float* __restrict__ logits,
                         float* __restrict__ a, float* __restrict__ ux,
                         float* __restrict__ uy, float* __restrict__ L) {
    int i = blockIdx.x * blockDim.x + threadIdx.x;
    if (i >= BATCH * NPTS) return;
    float x = dp[2 * i], y = dp[2 * i + 1];
    float r2 = x * x + y * y;
    float inv = rsqrtf(r2);
    a[i]  = __expf(-0.5f * r2);
    ux[i] = x * inv;
    uy[i] = y * inv;
    L[i]  = logits[i];
}

// ---- Wsym = (W + W^T)/2, 32x32 LDS tile transpose (one-time) ----------------
__global__ void crf_sym(const float* __restrict__ W, float* __restrict__ Ws) {
    __shared__ float tA[32][33];
    __shared__ float tB[32][33];
    int j0 = blockIdx.x * 32, i0 = blockIdx.y * 32;
    int c = threadIdx.x;
    for (int rr = 0; rr < 32; rr += 8) {
        int r = threadIdx.y + rr;
        tA[r][c] = W[(size_t)(i0 + r) * NPTS + (j0 + c)];
        tB[r][c] = W[(size_t)(j0 + r) * NPTS + (i0 + c)];
    }
    __syncthreads();
    for (int rr = 0; rr < 32; rr += 8) {
        int r = threadIdx.y + rr;
        Ws[(size_t)(i0 + r) * NPTS + (j0 + c)] = 0.5f * (tA[r][c] + tB[c][r]);
    }
}

// ---- combine split-K partials for one (batch, point) ------------------------
__device__ __forceinline__ float crf_combine(const float* __restrict__ Yp,
                                             int p, int b, float uxv, float uyv) {
    float y0 = 0.f, y1 = 0.f, y2 = 0.f;
#pragma unroll
    for (int s = 0; s < SPLITK; ++s) {
        const float* Ys = Yp + (size_t)s * NPTS * NC + (size_t)p * NC + 3 * b;
        y0 += Ys[0];
        y1 += Ys[1];
        y2 += Ys[2];
    }
    return y0 - uxv * y1 - uyv * y2;
}

// ---- fuse logit update (from previous Y partials) with building V -----------
__global__ void crf_pointv(const float* __restrict__ a, const float* __restrict__ ux,
                           const float* __restrict__ uy, const float* __restrict__ unary,
                           const float* __restrict__ Yp, float* __restrict__ L,
                           float* __restrict__ V, int useY) {
    int i = blockIdx.x * blockDim.x + threadIdx.x;
    if (i >= BATCH * NPTS) return;
    int b = i >> 12;          // i / NPTS
    int p = i & (NPTS - 1);   // i % NPTS
    float uxv = ux[i], uyv = uy[i], av = a[i];
    float l;
    if (useY) {
        l = unary[i] + av * crf_combine(Yp, p, b, uxv, uyv);
    } else {
        l = L[i];
    }
    L[i] = l;
    float q = 2.0f / (1.0f + __expf(-l)) - 1.0f;   // 2*sigmoid(l)-1
    float s = av * q;
    V[p * NC + 3 * b]     = s;
    V[p * NC + 3 * b + 1] = s * uxv;
    V[p * NC + 3 * b + 2] = s * uyv;
    if (b == 0) {
#pragma unroll
        for (int cc = 24; cc < 32; ++cc) V[p * NC + cc] = 0.0f;  // pad cols
    }
}

// ---- skinny GEMM: Ypart[s](4096x32) = Wsym[:, s-slice] @ V[s-slice, :] ------
// wave = one 16-row slab x both 16-col tiles; block = 8 row slabs at one split.
__global__ __launch_bounds__(256)
void crf_gemm(const float* __restrict__ Ws, const float* __restrict__ V,
              float* __restrict__ Ypart) {
    // K-pair interleaved: ldsV[(k>>1)*PSTR + col*2 + (k&1)]
    __shared__ float ldsV[(KC / 2) * PSTR];

    int tid   = threadIdx.x;
    int lane  = tid & 31;
    int wid   = tid >> 5;
    int split = blockIdx.x & (SPLITK - 1);      // K slice
    int rtg   = blockIdx.x >> 2;                // row-tile group 0..31
    int row0  = (rtg * 8 + wid) * 16;           // M tile base
    int kbeg  = split * KSPAN;

    int mrow = row0 + (lane & 15);
    int koff = (lane >> 4) << 1;                // 0 | 2
    const float* pA = Ws + (size_t)mrow * NPTS + koff;
    int nn = lane & 15;
    // per-lane base into LDS: pair offset from koff, column offset from nn
    int pbOff = (koff >> 1) * PSTR + nn * 2;

    v8f acc0 = {0.f, 0.f, 0.f, 0.f, 0.f, 0.f, 0.f, 0.f};
    v8f acc1 = {0.f, 0.f, 0.f, 0.f, 0.f, 0.f, 0.f, 0.f};

    for (int k0 = kbeg; k0 < kbeg + KSPAN; k0 += KC) {
        // cooperative stage of V[k0:k0+KC, 0:32] into LDS (coalesced global)
        const float* Vg = V + (size_t)k0 * NC;
        for (int idx = tid; idx < KC * NC; idx += 256) {
            int kk = idx >> 5, cc = idx & 31;
            ldsV[(kk >> 1) * PSTR + cc * 2 + (kk & 1)] = Vg[idx];
        }
        __syncthreads();

#pragma unroll 4
        for (int kk = 0; kk < KC; kk += 4) {
            // A fragment 16x4 f32: lane<16 -> K=kk,kk+1 ; lane>=16 -> K=kk+2,kk+3
            v2f av = *(const v2f*)(pA + k0 + kk);
            // B fragments: rows (kb,kb+1) at col nn / nn+16 are contiguous pairs
            const float* pb = &ldsV[(kk >> 1) * PSTR + pbOff];
            v2f b0 = *(const v2f*)(pb);
            v2f b1 = *(const v2f*)(pb + 32);
            acc0 = __builtin_amdgcn_wmma_f32_16x16x4_f32(
                false, av, false, b0, (short)0, acc0, false, false);
            acc1 = __builtin_amdgcn_wmma_f32_16x16x4_f32(
                false, av, false, b1, (short)0, acc1, false, false);
        }
        __syncthreads();
    }

    // C/D layout: vgpr r, lanes 0-15 -> M=r, lanes 16-31 -> M=r+8, N=lane&15
    float* Yp = Ypart + (size_t)split * NPTS * NC;
    int mbase = row0 + ((lane >> 4) << 3);
#pragma unroll
    for (int r = 0; r < 8; ++r) {
        Yp[(size_t)(mbase + r) * NC + nn]      = acc0[r];
        Yp[(size_t)(mbase + r) * NC + 16 + nn] = acc1[r];
    }
}

// ---- final: out = unary + a*(combined Y partials) ---------------------------
__global__ void crf_final(const float* __restrict__ a, const float* __restrict__ ux,
                          const float* __restrict__ uy, const float* __restrict__ unary,
                          const float* __restrict__ Yp, float* __restrict__ out) {
    int i = blockIdx.x * blockDim.x + threadIdx.x;
    if (i >= BATCH * NPTS) return;
    int b = i >> 12;
    int p = i & (NPTS - 1);
    out[i] = unary[i] + a[i] * crf_combine(Yp, p, b, ux[i], uy[i]);
}

extern "C" void kernel_launch(void* const* d_in, const int* in_sizes, int n_in,
                              void* d_out, int out_size, void* d_ws, size_t ws_size,
                              hipStream_t stream) {
    const float* dp     = (const float*)d_in[0];   // (8,64,64,2)
    const float* logits = (const float*)d_in[1];   // (8,4096,1)
    const float* W      = (const float*)d_in[2];   // (1,4096,4096)
    float* out = (float*)d_out;

    float* ws    = (float*)d_ws;
    float* Wsym  = ws;                                    // 16M floats (64 MB)
    float* V     = Wsym + (size_t)NPTS * NPTS;            // 4096*32
    float* Ypart = V + (size_t)NPTS * NC;                 // 4*4096*32 (2 MB)
    float* a     = Ypart + (size_t)SPLITK * NPTS * NC;    // 8*4096
    float* ux    = a + BATCH * NPTS;
    float* uy    = ux + BATCH * NPTS;
    float* L     = uy + BATCH * NPTS;

    crf_prep<<<128, 256, 0, stream>>>(dp, logits, a, ux, uy, L);
    crf_sym<<<dim3(128, 128), dim3(32, 8), 0, stream>>>(W, Wsym);

    for (int it = 0; it < ITERS; ++it) {
        crf_pointv<<<128, 256, 0, stream>>>(a, ux, uy, logits, Ypart, L, V, it);
        crf_gemm<<<32 * SPLITK, 256, 0, stream>>>(Wsym, V, Ypart);
    }
    crf_final<<<128, 256, 0, stream>>>(a, ux, uy, logits, Ypart, out);
}